// NSVQ_17763984736624
// MI455X (gfx1250) — compile-verified
//
#include <hip/hip_runtime.h>
#include <math.h>

typedef __attribute__((ext_vector_type(2))) float v2f;
typedef __attribute__((ext_vector_type(8))) float v8f;

// gfx1250 async global->LDS copy (ASYNCcnt-tracked), per-lane 16B.
// dst-LDS-address VGPR first, then 64-bit global address, saddr=off.
#define ASYNC_LOAD_B128(ldsaddr, gptr)                                  \
    asm volatile("global_load_async_to_lds_b128 %0, %1, off"            \
                 :: "v"(ldsaddr), "v"(gptr) : "memory")
#define WAIT_ASYNC() asm volatile("s_wait_asynccnt 0x0" ::: "memory")

// ---------------------------------------------------------------------------
// Generic WMMA f32 GEMM:  C[M,N] = op(A)[M,K] @ B[K,N] + bias[N]  (opt. relu)
// conv_mode=1: A is implicit im2col of convSrc[B][IH*IW][256] with 3x3 window.
// Block: 512 threads (16 waves, 4 per SIMD32). Block tile 128x256, K-chunk 32.
// Wave w owns a 16x128 sub-tile -> 8 v8f accumulators per wave.
// A tile staged via GLOBAL_LOAD_ASYNC_TO_LDS_B128 in direct mode (no VGPR
// round-trip; DMA overlaps the B-staging gather). Conv mode keeps the VGPR
// path because OOB taps must be zero-filled.
// B is staged k-paired: Bs[p][n] = {B[2p][n], B[2p+1][n]} so a WMMA B-frag is
// one aligned ds_load_b64 (row stride 544 floats == 32 mod 64 -> conflict-free).
// ---------------------------------------------------------------------------
__global__ __launch_bounds__(512)
void wmma_gemm_f32(const float* __restrict__ A, const float* __restrict__ Bm,
                   const float* __restrict__ bias, float* __restrict__ C,
                   int M, int N, int K, int relu,
                   int conv_mode, const float* __restrict__ convSrc,
                   int IH, int IW, int OH, int OW, int stride, int pad)
{
    __shared__ float As[128 * 36];   // 128 rows x 32 k, stride 36
    __shared__ float Bs[16 * 544];   // 16 k-pairs x 256 n (paired), stride 544

    const int tid  = threadIdx.x;
    const int wave = tid >> 5;       // 0..15
    const int lane = tid & 31;
    const int half = lane >> 4;      // 0: K=0,1  1: K=2,3 (A/B frag layout)
    const int lcol = lane & 15;
    const int m0 = blockIdx.x * 128;
    const int n0 = blockIdx.y * 256;
    const int mstrip = (wave >> 1) * 16;   // wave's 16-row strip
    const int nhalf  = (wave & 1) * 128;   // wave's 128-col half

    v8f acc[8];
#pragma unroll
    for (int i = 0; i < 8; ++i)
#pragma unroll
        for (int j = 0; j < 8; ++j) acc[i][j] = 0.f;

    for (int kk = 0; kk < K; kk += 32) {
        // ---- stage A tile (128 x 32) : 2 x 16B slots per thread ----
        if (!conv_mode) {
            // async DMA straight into LDS; overlaps with B staging below
#pragma unroll
            for (int i = 0; i < 2; ++i) {
                int fid = tid + i * 512;          // 0..1023
                int row = fid >> 3;               // 0..127
                int vec = fid & 7;                // float4 slot
                unsigned lds = (unsigned)(size_t)(As + row * 36 + vec * 4);
                const float* g = A + (size_t)(m0 + row) * K + kk + vec * 4;
                ASYNC_LOAD_B128(lds, g);
            }
        } else {
#pragma unroll
            for (int i = 0; i < 2; ++i) {
                int fid = tid + i * 512;
                int row = fid >> 3;
                int vec = fid & 7;
                int gm = m0 + row;
                int gk = kk + vec * 4;
                float4 val = make_float4(0.f, 0.f, 0.f, 0.f);
                int ohw = OH * OW;
                int b  = gm / ohw;
                int rr = gm - b * ohw;
                int oi = rr / OW;
                int oj = rr - oi * OW;
                int dd = gk >> 8;             // C = 256 fixed
                int c  = gk & 255;
                int di = dd / 3, dj = dd - di * 3;
                int si = oi * stride - pad + di;
                int sj = oj * stride - pad + dj;
                if (si >= 0 && si < IH && sj >= 0 && sj < IW)
                    val = *(const float4*)(convSrc +
                          ((size_t)(b * IH * IW + si * IW + sj)) * 256 + c);
                *(float4*)(As + row * 36 + vec * 4) = val;
            }
        }
        // ---- stage B tile (32 x 256) in k-paired layout ----
        // two threads per column: thread handles 8 k-pairs of column n0+(tid&255)
        {
            const int col = tid & 255;
            const int pbase = (tid >> 8) * 8;
            const float* bcol = Bm + (size_t)kk * N + n0 + col;
#pragma unroll
            for (int i = 0; i < 8; ++i) {
                int p = pbase + i;
                v2f bv;
                bv.x = bcol[(size_t)(2 * p) * N];
                bv.y = bcol[(size_t)(2 * p + 1) * N];
                *(v2f*)(Bs + p * 544 + col * 2) = bv;
            }
        }
        // prefetch next K-chunk while this one computes (global_prefetch_b8)
        if (kk + 32 < K) {
            __builtin_prefetch(Bm + (size_t)(kk + 32 + (tid >> 6)) * N + n0 + (tid & 63) * 4, 0, 1);
            if (!conv_mode)
                __builtin_prefetch(A + (size_t)(m0 + (tid >> 2)) * K + kk + 32 + (tid & 3) * 8, 0, 1);
        }
        if (!conv_mode) WAIT_ASYNC();     // our async A copies are in LDS
        __syncthreads();

        // ---- 8 k-steps x 8 n-tiles of V_WMMA_F32_16X16X4_F32 ----
#pragma unroll
        for (int k4 = 0; k4 < 32; k4 += 4) {
            v2f a = *(const v2f*)(As + (mstrip + lcol) * 36 + k4 + half * 2);
            const float* bbase = Bs + ((k4 >> 1) + half) * 544 + nhalf * 2;
#pragma unroll
            for (int nt = 0; nt < 8; ++nt) {
                v2f b = *(const v2f*)(bbase + (nt * 16 + lcol) * 2);
                acc[nt] = __builtin_amdgcn_wmma_f32_16x16x4_f32(
                    false, a, false, b, (short)0, acc[nt], false, false);
            }
        }
        __syncthreads();
    }

    // ---- epilogue: bias (+relu), store ----
#pragma unroll
    for (int nt = 0; nt < 8; ++nt) {
        int n = n0 + nhalf + nt * 16 + lcol;
        float bv = bias[n];
#pragma unroll
        for (int v = 0; v < 8; ++v) {
            int m = m0 + mstrip + v + half * 8;
            float r = acc[nt][v] + bv;
            if (relu) r = fmaxf(r, 0.f);
            C[(size_t)m * N + n] = r;
        }
    }
}

// ---------------------------------------------------------------------------
// VQ argmin via WMMA: score = x.c - 0.5*||c||^2 ; argmin dist == argmax score
// Grid: 64 blocks (one per 16-row group of x). 8 waves split the 8192 codes.
// ---------------------------------------------------------------------------
__global__ __launch_bounds__(256)
void vq_argmin_kernel(const float* __restrict__ x, const float* __restrict__ cb,
                      const float* __restrict__ cnorm_half, int* __restrict__ min_idx)
{
    __shared__ float Xs[16 * 264];
    __shared__ float Rs[2048];
    __shared__ int   Ri[2048];

    const int tid = threadIdx.x;
    const int g = blockIdx.x;

#pragma unroll
    for (int i = 0; i < 4; ++i) {
        int fid = tid + i * 256;
        int row = fid >> 6, c4 = fid & 63;
        *(float4*)(Xs + row * 264 + c4 * 4) =
            *(const float4*)(x + (size_t)(g * 16 + row) * 256 + c4 * 4);
    }
    __syncthreads();

    const int wave = tid >> 5, lane = tid & 31;
    const int half = lane >> 4, lcol = lane & 15;

    v2f af[64];
#pragma unroll
    for (int s = 0; s < 64; ++s)
        af[s] = *(const v2f*)(Xs + lcol * 264 + s * 4 + half * 2);

    float best[8];
    int   bidx[8];
#pragma unroll
    for (int v = 0; v < 8; ++v) { best[v] = -3.4e38f; bidx[v] = 0; }

    const int nbase = wave * 1024;
    for (int t = 0; t < 64; ++t) {
        int myn = nbase + t * 16 + lcol;
        const float* crow = cb + (size_t)myn * 256 + half * 2;
        v8f acc;
#pragma unroll
        for (int j = 0; j < 8; ++j) acc[j] = 0.f;
#pragma unroll
        for (int s = 0; s < 64; ++s) {
            v2f b = *(const v2f*)(crow + s * 4);
            acc = __builtin_amdgcn_wmma_f32_16x16x4_f32(
                false, af[s], false, b, (short)0, acc, false, false);
        }
        float hn = cnorm_half[myn];
#pragma unroll
        for (int v = 0; v < 8; ++v) {
            float sc = acc[v] - hn;
            if (sc > best[v] || (sc == best[v] && myn < bidx[v])) {
                best[v] = sc; bidx[v] = myn;
            }
        }
    }
    // per-lane partials -> LDS: entry (wave, row, col)
#pragma unroll
    for (int v = 0; v < 8; ++v) {
        int r = v + half * 8;
        Rs[wave * 256 + r * 16 + lcol] = best[v];
        Ri[wave * 256 + r * 16 + lcol] = bidx[v];
    }
    __syncthreads();
    {   // reduce across waves: thread t owns (row=t/16, col=t%16)
        int r = tid >> 4, j = tid & 15;
        float bs = Rs[r * 16 + j]; int bi = Ri[r * 16 + j];
        for (int w = 1; w < 8; ++w) {
            float s = Rs[w * 256 + r * 16 + j]; int ii = Ri[w * 256 + r * 16 + j];
            if (s > bs || (s == bs && ii < bi)) { bs = s; bi = ii; }
        }
        Rs[r * 16 + j] = bs; Ri[r * 16 + j] = bi;
    }
    __syncthreads();
    if (tid < 16) {   // reduce across the 16 columns of each row
        float bs = Rs[tid * 16]; int bi = Ri[tid * 16];
        for (int j = 1; j < 16; ++j) {
            float s = Rs[tid * 16 + j]; int ii = Ri[tid * 16 + j];
            if (s > bs || (s == bs && ii < bi)) { bs = s; bi = ii; }
        }
        min_idx[g * 16 + tid] = bi;
    }
}

// ---------------------------------------------------------------------------
// Per-row finalize: q = x + (||x-hard||/||rnd|| + eps) * rnd, written in the
// reference's reshape(B,EMB,4)->transpose permutation as decode-A. Also counts.
// ---------------------------------------------------------------------------
__global__ __launch_bounds__(256)
void vq_finalize_kernel(const float* __restrict__ x, const float* __restrict__ cb,
                        const float* __restrict__ rnd, const int* __restrict__ min_idx,
                        int* __restrict__ counts, int* __restrict__ min_idx_out,
                        float* __restrict__ A_dec)
{
    __shared__ float s1[256], s2[256];
    const int r = blockIdx.x, c = threadIdx.x;
    const int idx = min_idx[r];
    float xc = x[(size_t)r * 256 + c];
    float hc = cb[(size_t)idx * 256 + c];
    float rc = rnd[(size_t)r * 256 + c];
    float d = xc - hc;
    s1[c] = d * d;
    s2[c] = rc * rc;
    __syncthreads();
    for (int off = 128; off > 0; off >>= 1) {
        if (c < off) { s1[c] += s1[c + off]; s2[c] += s2[c + off]; }
        __syncthreads();
    }
    float scale = sqrtf(s1[0]) / sqrtf(s2[0]) + 1e-12f;
    float q = xc + scale * rc;
    // q[r][c] -> A_dec[b*4 + (c%4)][ (r%4)*64 + c/4 ]   (b = r/4)
    int b = r >> 2, rp = r & 3;
    A_dec[(size_t)(b * 4 + (c & 3)) * 256 + (rp * 64 + (c >> 2))] = q;
    if (c == 0) {
        atomicAdd(counts + idx, 1);
        min_idx_out[r] = idx;
    }
}

__global__ __launch_bounds__(256)
void perplexity_kernel(const int* __restrict__ counts, float* __restrict__ out_perp,
                       int* __restrict__ counts_out)
{
    __shared__ float s[256];
    const int tid = threadIdx.x;
    float local = 0.f;
    for (int i = tid; i < 8192; i += 256) {
        int cnt = counts[i];
        counts_out[i] = cnt;
        float avg = (float)cnt * (1.0f / 1024.0f);
        local += avg * logf(avg + 1e-12f);
    }
    s[tid] = local;
    __syncthreads();
    for (int off = 128; off > 0; off >>= 1) {
        if (tid < off) s[tid] += s[tid + off];
        __syncthreads();
    }
    if (tid == 0) out_perp[0] = expf(-s[0]);
}

__global__ __launch_bounds__(256)
void diff_kernel(const float* __restrict__ a, const float* __restrict__ b,
                 float* __restrict__ out, int n)
{
    int i = blockIdx.x * 256 + threadIdx.x;
    if (i < n) out[i] = a[i] - b[i];
}

__global__ __launch_bounds__(256)
void cnorm_kernel(const float* __restrict__ cb, float* __restrict__ cn)
{
    int n = blockIdx.x * 256 + threadIdx.x;   // 8192 codebooks
    const float4* p = (const float4*)(cb + (size_t)n * 256);
    float s = 0.f;
#pragma unroll 4
    for (int i = 0; i < 64; ++i) {
        float4 v = p[i];
        s += v.x * v.x + v.y * v.y + v.z * v.z + v.w * v.w;
    }
    cn[n] = 0.5f * s;
}

__global__ __launch_bounds__(256)
void zero_int_kernel(int* __restrict__ p, int n)
{
    int i = blockIdx.x * 256 + threadIdx.x;
    if (i < n) p[i] = 0;
}

// conv weight [O=256][C=256][3][3] -> B[k=(di*3+dj)*256+c][o]
__global__ __launch_bounds__(256)
void reorder_w_kernel(const float* __restrict__ w, float* __restrict__ Bm)
{
    int k = blockIdx.x;        // 0..2303
    int o = threadIdx.x;       // 0..255
    int dd = k >> 8, c = k & 255;
    Bm[(size_t)k * 256 + o] = w[(size_t)o * 2304 + c * 9 + dd];
}

extern "C" void kernel_launch(void* const* d_in, const int* in_sizes, int n_in,
                              void* d_out, int out_size, void* d_ws, size_t ws_size,
                              hipStream_t stream)
{
    const float* in_f  = (const float*)d_in[0];
    const float* in_l  = (const float*)d_in[1];
    const float* rnd   = (const float*)d_in[2];
    const float* cb    = (const float*)d_in[3];
    const float* W_in  = (const float*)d_in[4];
    const float* b_in  = (const float*)d_in[5];
    const float* W_out = (const float*)d_in[6];
    const float* b_out = (const float*)d_in[7];
    const float* c1w   = (const float*)d_in[8];
    const float* c1b   = (const float*)d_in[9];
    const float* c2w   = (const float*)d_in[10];
    const float* c2b   = (const float*)d_in[11];
    float* out = (float*)d_out;

    // workspace carve-up (floats)
    float* ws = (float*)d_ws;
    size_t off = 0;
    auto alloc = [&](size_t n) { float* p = ws + off; off += n; return p; };
    float* y_f  = alloc((size_t)16384 * 256);
    float* y_l  = alloc((size_t)16384 * 256);
    float* c1f  = alloc((size_t)4096 * 256);
    float* c1l  = alloc((size_t)4096 * 256);
    float* c2f  = alloc((size_t)1024 * 256);
    float* c2l  = alloc((size_t)1024 * 256);
    float* xbuf = alloc((size_t)1024 * 256);
    float* Adec = alloc((size_t)1024 * 256);
    float* Bc1  = alloc((size_t)2304 * 256);
    float* Bc2  = alloc((size_t)2304 * 256);
    float* cn   = alloc(8192);
    int*   counts = (int*)alloc(8192);
    int*   midx   = (int*)alloc(1024);

    // output layout: dec[1048576] | perplexity[1] | codebooks_used[8192] | min_indices[1024]
    float* dec        = out;
    float* perp       = out + 1048576;
    int*   counts_out = (int*)(out + 1048577);
    int*   midx_out   = (int*)(out + 1048577 + 8192);

    // prep
    reorder_w_kernel<<<2304, 256, 0, stream>>>(c1w, Bc1);
    reorder_w_kernel<<<2304, 256, 0, stream>>>(c2w, Bc2);
    cnorm_kernel<<<32, 256, 0, stream>>>(cb, cn);
    zero_int_kernel<<<32, 256, 0, stream>>>(counts, 8192);

    // project_in (both inputs): [16384,1024] @ [1024,256] + b_in
    wmma_gemm_f32<<<dim3(128, 1), 512, 0, stream>>>(
        in_f, W_in, b_in, y_f, 16384, 256, 1024, 0, 0, nullptr, 0, 0, 0, 0, 0, 0);
    wmma_gemm_f32<<<dim3(128, 1), 512, 0, stream>>>(
        in_l, W_in, b_in, y_l, 16384, 256, 1024, 0, 0, nullptr, 0, 0, 0, 0, 0, 0);

    // conv1 (3x3 s2 p1, 8x8->4x4) as implicit-im2col GEMM, +bias, relu
    wmma_gemm_f32<<<dim3(32, 1), 512, 0, stream>>>(
        nullptr, Bc1, c1b, c1f, 4096, 256, 2304, 1, 1, y_f, 8, 8, 4, 4, 2, 1);
    wmma_gemm_f32<<<dim3(32, 1), 512, 0, stream>>>(
        nullptr, Bc1, c1b, c1l, 4096, 256, 2304, 1, 1, y_l, 8, 8, 4, 4, 2, 1);

    // conv2 (3x3 s1 p0, 4x4->2x2), +bias
    wmma_gemm_f32<<<dim3(8, 1), 512, 0, stream>>>(
        nullptr, Bc2, c2b, c2f, 1024, 256, 2304, 0, 1, c1f, 4, 4, 2, 2, 1, 0);
    wmma_gemm_f32<<<dim3(8, 1), 512, 0, stream>>>(
        nullptr, Bc2, c2b, c2l, 1024, 256, 2304, 0, 1, c1l, 4, 4, 2, 2, 1, 0);

    // x = last - first
    diff_kernel<<<1024, 256, 0, stream>>>(c2l, c2f, xbuf, 1024 * 256);

    // VQ argmin + finalize + stats
    vq_argmin_kernel<<<64, 256, 0, stream>>>(xbuf, cb, cn, midx);
    vq_finalize_kernel<<<1024, 256, 0, stream>>>(xbuf, cb, rnd, midx,
                                                 counts, midx_out, Adec);
    perplexity_kernel<<<1, 256, 0, stream>>>(counts, perp, counts_out);

    // decode: [1024,256] @ [256,1024] + b_out -> dec
    wmma_gemm_f32<<<dim3(8, 4), 512, 0, stream>>>(
        Adec, W_out, b_out, dec, 1024, 1024, 256, 0, 0, nullptr, 0, 0, 0, 0, 0, 0);
}